// EnhancedGNNTransformerEncoder_43465069035554
// MI455X (gfx1250) — compile-verified
//
#include <hip/hip_runtime.h>
#include <math.h>

// ---------------------------------------------------------------------------
// EnhancedGNNTransformerEncoder for MI455X (gfx1250, wave32, WMMA)
//
// Input order (setup_inputs() insertion order, nested dicts flattened in
// insertion order):
//   d_in[0]  x           [20000,256] f32
//   d_in[1]  edge_index  [2,320000]  i32   (row0=src, row1=dst)
//   d_in[2]  batch       [20000]     i32   (unused)
//   d_in[3+8l+0..7]  layer l: qW,qb,kW,kb,vW,vb,skW,skb
//   d_in[35],[36]    proj0 W,b
//   d_in[37],[38]    out   W,b
// ---------------------------------------------------------------------------

typedef __attribute__((ext_vector_type(2))) float v2f;
typedef __attribute__((ext_vector_type(8))) float v8f;

#define HC 128          // hidden*heads
#define HEADS 8
#define HDIM 16

__device__ __forceinline__ v8f wmma4(v2f a, v2f b, v8f c) {
  // D = A(16x4,f32) * B(4x16,f32) + C(16x16,f32)
  return __builtin_amdgcn_wmma_f32_16x16x4_f32(
      /*neg_a=*/false, a, /*neg_b=*/false, b,
      /*c_mod=*/(short)0, c, /*reuse_a=*/false, /*reuse_b=*/false);
}

// C[M,128] = A[M,K] @ W[K,128] + bias[128]
// block = 128 threads = 4 waves; wave -> one 16-row tile, 64 cols (grid.y picks half)
__global__ void __launch_bounds__(128)
gemm_bias_128(const float* __restrict__ A, const float* __restrict__ W,
              const float* __restrict__ bias, float* __restrict__ C,
              int M, int K) {
  const int lane = threadIdx.x & 31;
  const int wid  = threadIdx.x >> 5;
  const int mtiles = M >> 4;
  const int tileRow = blockIdx.x * 4 + wid;        // wave-uniform
  if (tileRow >= mtiles) return;                   // whole wave exits together
  const int m0 = tileRow << 4;
  const int n0 = blockIdx.y << 6;                  // 0 or 64
  const int lr   = lane & 15;
  const int half = lane >> 4;                      // 0: K pair {0,1}, 1: {2,3}
  const int koff = half << 1;

  const float* arow = A + (size_t)(m0 + lr) * K + koff;
  v8f c0 = {}, c1 = {}, c2 = {}, c3 = {};

  for (int k0 = 0; k0 < K; k0 += 4) {
    v2f a;
    a.x = arow[k0];
    a.y = arow[k0 + 1];
    const float* brow = W + (size_t)(k0 + koff) * HC + n0 + lr;
    v2f b;
    b.x = brow[0];   b.y = brow[HC + 0];   c0 = wmma4(a, b, c0);
    b.x = brow[16];  b.y = brow[HC + 16];  c1 = wmma4(a, b, c1);
    b.x = brow[32];  b.y = brow[HC + 32];  c2 = wmma4(a, b, c2);
    b.x = brow[48];  b.y = brow[HC + 48];  c3 = wmma4(a, b, c3);
  }

  // C/D layout: VGPR r -> row m0 + r + half*8; col n0 + lr (+16j)
  const int col = n0 + lr;
  const int r0  = m0 + (half << 3);
  const float b0 = bias[col], b1 = bias[col + 16],
              b2 = bias[col + 32], b3 = bias[col + 48];
#pragma unroll
  for (int r = 0; r < 8; ++r) {
    float* crow = C + (size_t)(r0 + r) * HC;
    crow[col]      = c0[r] + b0;
    crow[col + 16] = c1[r] + b1;
    crow[col + 32] = c2[r] + b2;
    crow[col + 48] = c3[r] + b3;
  }
}

// ---- monotonic uint encoding of float for atomic max --------------------
__device__ __forceinline__ unsigned encodeF(float f) {
  unsigned u = __float_as_uint(f);
  return (u & 0x80000000u) ? ~u : (u | 0x80000000u);
}
__device__ __forceinline__ float decodeF(unsigned u) {
  return __uint_as_float((u & 0x80000000u) ? (u ^ 0x80000000u) : ~u);
}

__global__ void layer_init(float* __restrict__ agg, unsigned* __restrict__ amax,
                           float* __restrict__ denom, int N) {
  int gid = blockIdx.x * blockDim.x + threadIdx.x;
  if (gid < N * HC) agg[gid] = 0.f;
  if (gid < N * HEADS) { amax[gid] = 0u; denom[gid] = 0.f; }
}

// alpha[e,h] = dot(q[dst,h,:], k[src,h,:]) / 4 ; amax[dst,h] = max(...)
__global__ void edge_logits(const int* __restrict__ ei,
                            const float* __restrict__ q,
                            const float* __restrict__ k,
                            float* __restrict__ alpha,
                            unsigned* __restrict__ amax, int E) {
  int gid = blockIdx.x * blockDim.x + threadIdx.x;
  if (gid >= E * HEADS) return;
  int e = gid >> 3, h = gid & 7;
  int src = ei[e], dst = ei[E + e];
  const float4* qp = (const float4*)(q + (size_t)dst * HC + h * HDIM);
  const float4* kp = (const float4*)(k + (size_t)src * HC + h * HDIM);
  float s = 0.f;
#pragma unroll
  for (int i = 0; i < 4; ++i) {
    float4 a = qp[i], b = kp[i];
    s += a.x * b.x + a.y * b.y + a.z * b.z + a.w * b.w;
  }
  s *= 0.25f;  // 1/sqrt(16)
  alpha[gid] = s;
  atomicMax(amax + (size_t)dst * HEADS + h, encodeF(s));
}

// alpha[e,h] = exp(alpha - amax[dst,h]) ; denom[dst,h] += alpha
__global__ void edge_exp(const int* __restrict__ ei,
                         const unsigned* __restrict__ amax,
                         float* __restrict__ alpha,
                         float* __restrict__ denom, int E) {
  int gid = blockIdx.x * blockDim.x + threadIdx.x;
  if (gid >= E * HEADS) return;
  int e = gid >> 3, h = gid & 7;
  int dst = ei[E + e];
  float m  = decodeF(amax[(size_t)dst * HEADS + h]);
  float ex = expf(alpha[gid] - m);
  alpha[gid] = ex;
  atomicAdd(denom + (size_t)dst * HEADS + h, ex);
}

// agg[dst,d] += v[src,d] * alpha[e,h]/denom[dst,h], one thread per (e,d)
__global__ void edge_scatter(const int* __restrict__ ei,
                             const float* __restrict__ alpha,
                             const float* __restrict__ denom,
                             const float* __restrict__ v,
                             float* __restrict__ agg, int E) {
  int gid = blockIdx.x * blockDim.x + threadIdx.x;
  if (gid >= E * HC) return;
  int e = gid >> 7, d = gid & 127, h = d >> 4;
  int src = ei[e], dst = ei[E + e];
  float w = alpha[(size_t)e * HEADS + h] / denom[(size_t)dst * HEADS + h];
  atomicAdd(agg + (size_t)dst * HC + d, v[(size_t)src * HC + d] * w);
}

// xout = relu(agg + sk) + res
__global__ void node_update(const float* __restrict__ agg,
                            const float* __restrict__ sk,
                            const float* __restrict__ res,
                            float* __restrict__ xout, int n) {
  int gid = blockIdx.x * blockDim.x + threadIdx.x;
  if (gid >= n) return;
  float t = agg[gid] + sk[gid];
  t = t > 0.f ? t : 0.f;
  xout[gid] = t + res[gid];
}

extern "C" void kernel_launch(void* const* d_in, const int* in_sizes, int n_in,
                              void* d_out, int out_size, void* d_ws, size_t ws_size,
                              hipStream_t stream) {
  const float* x  = (const float*)d_in[0];
  const int*   ei = (const int*)d_in[1];
  const int N = in_sizes[2];              // 20000
  const int E = in_sizes[1] / 2;          // 320000
  const int IN_CH = in_sizes[0] / N;      // 256

  auto P = [&](int i) { return (const float*)d_in[i]; };
  const float* proj0W = P(35); const float* proj0b = P(36);
  const float* outW   = P(37); const float* outb   = P(38);

  // ---- workspace carve-up (floats) ----
  const size_t NH = (size_t)N * HC;       // 2.56M
  float* ws   = (float*)d_ws;
  float* xA   = ws;             // layer outputs (ping)
  float* xB   = xA + NH;        // layer outputs (pong) / layer-0 residual
  float* q    = xB + NH;
  float* k    = q + NH;
  float* v    = k + NH;
  float* sk   = v + NH;
  float* agg  = sk + NH;
  float* alpha = agg + NH;                         // E*HEADS floats
  unsigned* amax = (unsigned*)(alpha + (size_t)E * HEADS);  // N*HEADS
  float* denom   = (float*)(amax + (size_t)N * HEADS);      // N*HEADS

  auto gemm = [&](const float* A, const float* W, const float* b, float* C,
                  int M, int K) {
    dim3 grid((M / 16 + 3) / 4, 2);
    gemm_bias_128<<<grid, 128, 0, stream>>>(A, W, b, C, M, K);
  };

  const int TB = 256;
  const int gEH = (E * HEADS + TB - 1) / TB;
  const int gEC = (E * HC + TB - 1) / TB;
  const int gNC = (N * HC + TB - 1) / TB;

  const float* xin = x;
  int Kin = IN_CH;
  for (int l = 0; l < 4; ++l) {
    const int base = 3 + 8 * l;
    gemm(xin, P(base + 0), P(base + 1), q,  N, Kin);
    gemm(xin, P(base + 2), P(base + 3), k,  N, Kin);
    gemm(xin, P(base + 4), P(base + 5), v,  N, Kin);
    gemm(xin, P(base + 6), P(base + 7), sk, N, Kin);

    const float* res;
    float* xout;
    if (l == 0) {
      gemm(xin, proj0W, proj0b, xB, N, Kin);   // residual projection
      res = xB; xout = xA;
    } else {
      res = xin;
      xout = (xin == xA) ? xB : xA;
    }

    layer_init<<<gNC, TB, 0, stream>>>(agg, amax, denom, N);
    edge_logits<<<gEH, TB, 0, stream>>>(ei, q, k, alpha, amax, E);
    edge_exp<<<gEH, TB, 0, stream>>>(ei, amax, alpha, denom, E);
    edge_scatter<<<gEC, TB, 0, stream>>>(ei, alpha, denom, v, agg, E);
    node_update<<<gNC, TB, 0, stream>>>(agg, sk, res, xout, N * HC);

    xin = xout;
    Kin = HC;
  }

  // final projection -> d_out [N,128]
  gemm(xin, outW, outb, (float*)d_out, N, HC);
}